// GIN_80247168958681
// MI455X (gfx1250) — compile-verified
//
#include <hip/hip_runtime.h>
#include <hip/hip_bf16.h>
#include <math.h>

#define N_NODES   20000
#define N_EDGES0  320000
#define E_TOT     (N_EDGES0 + N_NODES)
#define NUM_GRAPHS 64
#define HIDC      64
#define LATENT    32
#define NEG_SLOPE 0.2f
#define BN_EPS    1e-5f

typedef __attribute__((ext_vector_type(2))) float v2f;
typedef __attribute__((ext_vector_type(8))) float v8f;

// ---------------------------------------------------------------- utilities
__global__ void fill_kernel(float* __restrict__ p, float v, int n) {
  int i = blockIdx.x * blockDim.x + threadIdx.x;
  if (i < n) p[i] = v;
}

// ---------------------------------------------------------------- f32 WMMA GEMM
// C[N x M] = A[N x K] * B[K x M], all row-major f32.
// Requirements: N % 16 == 0, K % 4 == 0, M % 64 == 0.
// One wave computes a 16x64 strip (4x v_wmma_f32_16x16x4_f32 per K-step,
// sharing the A fragment). 8 waves per 256-thread block cover 8 row tiles.
__global__ __launch_bounds__(256) void gemm_wmma_f32(const float* __restrict__ A,
                                                     const float* __restrict__ B,
                                                     float* __restrict__ C,
                                                     int N, int K, int M) {
  const int lane = threadIdx.x & 31;
  const int wave = threadIdx.x >> 5;
  const int lo   = lane & 15;
  const int hi   = lane >> 4;          // 0: lanes 0-15, 1: lanes 16-31
  const int rowTile = blockIdx.x * 8 + wave;
  if (rowTile * 16 >= N) return;
  const int row     = rowTile * 16 + lo;
  const int colBase = blockIdx.y * 64;

  v8f c0 = {}, c1 = {}, c2 = {}, c3 = {};
  for (int kk = 0; kk < K; kk += 4) {
    // A fragment (16x4): lanes 0-15 hold K = kk..kk+1, lanes 16-31 K = kk+2..kk+3
    const int ka = kk + 2 * hi;
    v2f a;
    a.x = A[(size_t)row * K + ka];
    a.y = A[(size_t)row * K + ka + 1];
    // B fragments (4x16 each), same K striping as A
    const float* Brow0 = B + (size_t)ka * M + colBase + lo;
    const float* Brow1 = Brow0 + M;
    v2f b0; b0.x = Brow0[0];  b0.y = Brow1[0];
    v2f b1; b1.x = Brow0[16]; b1.y = Brow1[16];
    v2f b2; b2.x = Brow0[32]; b2.y = Brow1[32];
    v2f b3; b3.x = Brow0[48]; b3.y = Brow1[48];
    c0 = __builtin_amdgcn_wmma_f32_16x16x4_f32(false, a, false, b0, (short)0, c0, false, false);
    c1 = __builtin_amdgcn_wmma_f32_16x16x4_f32(false, a, false, b1, (short)0, c1, false, false);
    c2 = __builtin_amdgcn_wmma_f32_16x16x4_f32(false, a, false, b2, (short)0, c2, false, false);
    c3 = __builtin_amdgcn_wmma_f32_16x16x4_f32(false, a, false, b3, (short)0, c3, false, false);
  }
  // C/D layout: VGPR v -> row M=v (lanes 0-15) / M=v+8 (lanes 16-31)
#pragma unroll
  for (int v = 0; v < 8; ++v) {
    const size_t r = (size_t)(rowTile * 16 + v + 8 * hi) * M + colBase + lo;
    C[r +  0] = c0[v];
    C[r + 16] = c1[v];
    C[r + 32] = c2[v];
    C[r + 48] = c3[v];
  }
}

// ---------------------------------------------------------------- edge pass A
// logits[e][h] = sum_c att[h][c] * leaky_relu(xl[src][h][c] + xr[dst][h][c])
// plus segment-max over dst via float atomicMax. One wave per edge.
template <int H, int C>
__global__ __launch_bounds__(256) void edge_logits_kernel(const int* __restrict__ ei,
                                                          const float* __restrict__ xl,
                                                          const float* __restrict__ xr,
                                                          const float* __restrict__ att,
                                                          float* __restrict__ logits,
                                                          float* __restrict__ mmax) {
  constexpr int HC  = H * C;
  constexpr int PER = HC / 32;        // channels per lane
  constexpr int LPH = 32 / H;         // lanes per head
  const int lane = threadIdx.x & 31;
  const int wave = threadIdx.x >> 5;
  const int e = blockIdx.x * 8 + wave;
  if (e >= E_TOT) return;
  int s, d;
  if (e < N_EDGES0) { s = ei[e]; d = ei[N_EDGES0 + e]; }
  else              { s = e - N_EDGES0; d = s; }

  const float* pl = xl  + (size_t)s * HC + lane * PER;
  const float* pr = xr  + (size_t)d * HC + lane * PER;
  const float* pa = att + lane * PER;
  float partial = 0.f;
#pragma unroll
  for (int i = 0; i < PER; ++i) {
    float v = pl[i] + pr[i];
    v = v > 0.f ? v : NEG_SLOPE * v;
    partial += pa[i] * v;
  }
#pragma unroll
  for (int m = LPH >> 1; m >= 1; m >>= 1)
    partial += __shfl_xor(partial, m, 32);
  if ((lane % LPH) == 0) {
    const int h = (lane * PER) / C;
    logits[(size_t)e * H + h] = partial;
    atomicMax(&mmax[(size_t)d * H + h], partial);
  }
}

// ---------------------------------------------------------------- edge pass B
// ex = exp(logit - max); denom[dst][h] += ex; acc[dst][h][c] += ex * xl[src][h][c]
template <int H, int C>
__global__ __launch_bounds__(256) void edge_scatter_kernel(const int* __restrict__ ei,
                                                           const float* __restrict__ xl,
                                                           const float* __restrict__ logits,
                                                           const float* __restrict__ mmax,
                                                           float* __restrict__ denom,
                                                           float* __restrict__ acc) {
  constexpr int HC  = H * C;
  constexpr int PER = HC / 32;
  constexpr int LPH = 32 / H;
  const int lane = threadIdx.x & 31;
  const int wave = threadIdx.x >> 5;
  const int e = blockIdx.x * 8 + wave;
  if (e >= E_TOT) return;
  int s, d;
  if (e < N_EDGES0) { s = ei[e]; d = ei[N_EDGES0 + e]; }
  else              { s = e - N_EDGES0; d = s; }

  const int h = (lane * PER) / C;
  const float ex = expf(logits[(size_t)e * H + h] - mmax[(size_t)d * H + h]);
  if ((lane % LPH) == 0) atomicAdd(&denom[(size_t)d * H + h], ex);

  const float* pl = xl + (size_t)s * HC + lane * PER;
  float* pa = acc + (size_t)d * HC + lane * PER;
#pragma unroll
  for (int i = 0; i < PER; ++i) atomicAdd(&pa[i], ex * pl[i]);
}

// ---------------------------------------------------------------- finalize node
template <int H, int C>
__global__ void finalize_kernel(const float* __restrict__ acc,
                                const float* __restrict__ denom,
                                const float* __restrict__ bias,
                                float* __restrict__ out) {
  constexpr int HC = H * C;
  int i = blockIdx.x * blockDim.x + threadIdx.x;
  if (i >= N_NODES * HC) return;
  const int node = i / HC;
  const int ch   = i % HC;
  const int h    = ch / C;
  out[i] = acc[i] / denom[(size_t)node * H + h] + bias[ch];
}

// ---------------------------------------------------------------- pooling + head
__global__ void pool_kernel(const float* __restrict__ h,
                            const int* __restrict__ batch,
                            float* __restrict__ pooled) {
  int i = blockIdx.x * blockDim.x + threadIdx.x;
  if (i >= N_NODES * HIDC) return;
  const int node = i >> 6;
  const int c    = i & 63;
  atomicAdd(&pooled[(size_t)batch[node] * HIDC + c], h[i]);
}

__global__ void head_kernel(const float* __restrict__ pooled,
                            const float* __restrict__ gamma,
                            const float* __restrict__ beta,
                            const float* __restrict__ mean,
                            const float* __restrict__ var,
                            const float* __restrict__ fw,
                            const float* __restrict__ fb,
                            float* __restrict__ out) {
  int i = blockIdx.x * blockDim.x + threadIdx.x;
  if (i >= NUM_GRAPHS * LATENT) return;
  const int g = i >> 5;
  const int j = i & 31;
  float acc = fb[j];
#pragma unroll 8
  for (int c = 0; c < HIDC; ++c) {
    float bn = (pooled[g * HIDC + c] - mean[c]) * rsqrtf(var[c] + BN_EPS) * gamma[c] + beta[c];
    acc += bn * fw[c * LATENT + j];
  }
  out[i] = acc;
}

// ---------------------------------------------------------------- launcher
extern "C" void kernel_launch(void* const* d_in, const int* in_sizes, int n_in,
                              void* d_out, int out_size, void* d_ws, size_t ws_size,
                              hipStream_t stream) {
  const float* x     = (const float*)d_in[0];
  const int*   ei    = (const int*)d_in[1];
  const int*   batch = (const int*)d_in[2];
  const float* Wl[3] = { (const float*)d_in[3], (const float*)d_in[7],  (const float*)d_in[11] };
  const float* Wr[3] = { (const float*)d_in[4], (const float*)d_in[8],  (const float*)d_in[12] };
  const float* at[3] = { (const float*)d_in[5], (const float*)d_in[9],  (const float*)d_in[13] };
  const float* bs[3] = { (const float*)d_in[6], (const float*)d_in[10], (const float*)d_in[14] };
  const float* bn_gamma = (const float*)d_in[15];
  const float* bn_beta  = (const float*)d_in[16];
  const float* bn_mean  = (const float*)d_in[17];
  const float* bn_var   = (const float*)d_in[18];
  const float* fc_w     = (const float*)d_in[19];
  const float* fc_b     = (const float*)d_in[20];
  float* out = (float*)d_out;

  // workspace layout (floats)
  float* ws = (float*)d_ws;
  const size_t NHC = (size_t)N_NODES * 256;
  float* buf_h   = ws;                 ws += NHC;                 // current node features
  float* buf_xl  = ws;                 ws += NHC;
  float* buf_xr  = ws;                 ws += NHC;
  float* buf_acc = ws;                 ws += NHC;
  float* buf_m   = ws;                 ws += (size_t)N_NODES * 4;
  float* buf_den = ws;                 ws += (size_t)N_NODES * 4;
  float* buf_lg  = ws;                 ws += (size_t)E_TOT * 4;
  float* pooled  = ws;                 ws += (size_t)NUM_GRAPHS * HIDC;

  const dim3 blk(256);
  const int  rowBlocks  = (N_NODES / 16 + 7) / 8;         // 157
  const int  edgeBlocks = (E_TOT + 7) / 8;                // 42500
  auto cdiv = [](size_t a, size_t b) { return (unsigned)((a + b - 1) / b); };

  const int Kin[3] = { 128, 256, 256 };
  const int Mout[3] = { 256, 256, 64 };
  const int Hh[3] = { 4, 4, 1 };

  for (int l = 0; l < 3; ++l) {
    const float* in_feat = (l == 0) ? x : buf_h;
    const int K = Kin[l], M = Mout[l], H = Hh[l];
    const size_t nHC = (size_t)N_NODES * M;

    // xl = in @ Wl, xr = in @ Wr  (f32 WMMA)
    dim3 ggrid(rowBlocks, M / 64);
    gemm_wmma_f32<<<ggrid, blk, 0, stream>>>(in_feat, Wl[l], buf_xl, N_NODES, K, M);
    gemm_wmma_f32<<<ggrid, blk, 0, stream>>>(in_feat, Wr[l], buf_xr, N_NODES, K, M);

    // init segment-max / denom / accumulator
    fill_kernel<<<cdiv((size_t)N_NODES * H, 256), blk, 0, stream>>>(buf_m,  -3.0e38f, N_NODES * H);
    fill_kernel<<<cdiv((size_t)N_NODES * H, 256), blk, 0, stream>>>(buf_den, 0.0f,    N_NODES * H);
    fill_kernel<<<cdiv(nHC, 256),                 blk, 0, stream>>>(buf_acc, 0.0f,    (int)nHC);

    if (H == 4) {
      edge_logits_kernel<4, 64><<<edgeBlocks, blk, 0, stream>>>(ei, buf_xl, buf_xr, at[l], buf_lg, buf_m);
      edge_scatter_kernel<4, 64><<<edgeBlocks, blk, 0, stream>>>(ei, buf_xl, buf_lg, buf_m, buf_den, buf_acc);
      finalize_kernel<4, 64><<<cdiv(nHC, 256), blk, 0, stream>>>(buf_acc, buf_den, bs[l], buf_h);
    } else {
      edge_logits_kernel<1, 64><<<edgeBlocks, blk, 0, stream>>>(ei, buf_xl, buf_xr, at[l], buf_lg, buf_m);
      edge_scatter_kernel<1, 64><<<edgeBlocks, blk, 0, stream>>>(ei, buf_xl, buf_lg, buf_m, buf_den, buf_acc);
      finalize_kernel<1, 64><<<cdiv(nHC, 256), blk, 0, stream>>>(buf_acc, buf_den, bs[l], buf_h);
    }
  }

  // global_add_pool -> BN (eval) -> fc
  fill_kernel<<<cdiv((size_t)NUM_GRAPHS * HIDC, 256), blk, 0, stream>>>(pooled, 0.0f, NUM_GRAPHS * HIDC);
  pool_kernel<<<cdiv((size_t)N_NODES * HIDC, 256), blk, 0, stream>>>(buf_h, batch, pooled);
  head_kernel<<<cdiv((size_t)NUM_GRAPHS * LATENT, 256), blk, 0, stream>>>(
      pooled, bn_gamma, bn_beta, bn_mean, bn_var, fc_w, fc_b, out);
}